// HamiltonianFFN_77704548319439
// MI455X (gfx1250) — compile-verified
//
#include <hip/hip_runtime.h>

// ---------------------------------------------------------------------------
// HamiltonianFFN on MI455X (gfx1250, wave32).
// All matrix algebra is 64x64 fp32 via V_WMMA_F32_16X16X4_F32.
// eigh-free: Newton-Schulz for Sigma^{+-1/2}, scaling-and-squaring expm.
// One workgroup (4 wave32) per (b,n) problem; 6 LDS matrix buffers,
// ~100KB of the 320KB WGP LDS.
//
// Codegen notes (round 3): LDS matrices are stored COLUMN-major
// ((i,j) -> j*ST+i, ST=65). This makes the WMMA B-fragment values
// (B[k][n], B[k+1][n]) adjacent in LDS, so each B pair is one
// ds_load_2addr_b32 straight into a consecutive VGPR pair (no v_mov
// re-pairing), with the k-displacement folded into the 8-bit immediate
// offsets (no in-loop address math). A-fragments (shared by 4 WMMAs per
// slab) take the strided path instead. All LDS traffic is rooted at a
// file-scope extern __shared__ array indexed by integer offsets to keep
// addrspace(3) 32-bit addressing.
// ---------------------------------------------------------------------------

typedef float v2f __attribute__((ext_vector_type(2)));
typedef float v8f __attribute__((ext_vector_type(8)));

#define KDIM      64
#define ST        65                 // LDS column stride in floats
#define IDX(i, j) ((j) * ST + (i))   // column-major element (row i, col j)
#define MAT_ELEMS (KDIM * ST)
#define NTHREADS  128                // 4 waves of 32
#define EPSV      1e-8f
#define DT2       0.2f               // dt * 2  (dt = 0.1)
#define NS_ITERS  6
#define BATCH     8

// Dynamic LDS pool shared by all kernels/helpers (keeps addrspace(3)).
extern __shared__ float lds[];

// lds[offC..] = alpha*(A@B) + diag*I for 64x64 column-major matrices at
// float offsets offA/offB/offC. Wave w computes rows 16w..16w+15, holding
// 4 accumulator tiles (4 x v8f) in VGPRs.
// fp32 WMMA fragment layout per CDNA5 ISA 7.12.2:
//   A 16x4 : lanes 0-15 M=0..15 {K=0,K=1}, lanes 16-31 M=0..15 {K=2,K=3}
//   B 4x16 : lanes 0-15 N=0..15 {K=0,K=1}, lanes 16-31 N=0..15 {K=2,K=3}
//   C 16x16: lane N = lane&15, vgpr r -> M = r + 8*(lane>>4)
__device__ __noinline__ void gemm64(int offA, int offB, int offC,
                                    float alpha, float diag)
{
    const int tid  = threadIdx.x;
    const int lane = tid & 31;
    const int wave = tid >> 5;
    const int half = lane >> 4;     // which K-pair this half-wave carries
    const int mr   = lane & 15;     // M for A-frag, N for B/C-frags
    const int arow = wave * 16 + mr;

    // Column-major bases. B/C: all in-loop displacements are small consts.
    const int abase = offA + arow + 2 * half * ST;       // + kk*ST, +ST
    int bbase[4];
#pragma unroll
    for (int j = 0; j < 4; ++j)
        bbase[j] = offB + (j * 16 + mr) * ST + 2 * half; // + kk, +1

    v8f zero = {0.f,0.f,0.f,0.f,0.f,0.f,0.f,0.f};
    v8f acc[4];
#pragma unroll
    for (int j = 0; j < 4; ++j) acc[j] = zero;

#pragma unroll
    for (int kk = 0; kk < KDIM; kk += 4) {
        // Batch all fragment loads for this k-slab, then issue 4 WMMAs.
        v2f a;
        a.x = lds[abase + kk * ST];       // A[arow][kk+2h]
        a.y = lds[abase + kk * ST + ST];  // A[arow][kk+2h+1]
        v2f b[4];
#pragma unroll
        for (int j = 0; j < 4; ++j) {
            b[j].x = lds[bbase[j] + kk];      // B[kk+2h  ][j*16+mr]
            b[j].y = lds[bbase[j] + kk + 1];  // B[kk+2h+1][j*16+mr]
        }
#pragma unroll
        for (int j = 0; j < 4; ++j) {
            acc[j] = __builtin_amdgcn_wmma_f32_16x16x4_f32(
                false, a, false, b[j], (short)0, acc[j], false, false);
        }
    }

    const int mbase = wave * 16 + half * 8;
#pragma unroll
    for (int j = 0; j < 4; ++j) {
        const int cbase = offC + (j * 16 + mr) * ST + mbase;
#pragma unroll
        for (int r = 0; r < 8; ++r) {
            float v = alpha * acc[j][r];
            if (mbase + r == j * 16 + mr) v += diag;   // + diag*I
            lds[cbase + r] = v;                        // C[mbase+r][j*16+mr]
        }
    }
    __syncthreads();   // publish C before anyone consumes it
}

// Global row-major 64x64 -> LDS column-major (transposing scatter).
// Global side stays float4-coalesced; LDS banks remain conflict-free
// because stride 65 keeps the lane->bank map a dense permutation.
__device__ inline void load_mat(const float* __restrict__ g, int dstoff)
{
    const float4* g4 = (const float4*)g;
    for (int q = threadIdx.x; q < (KDIM * KDIM / 4); q += NTHREADS) {
        float4 v = g4[q];
        const int i = q >> 4;            // row (16 float4 per row)
        const int j = (q & 15) << 2;     // starting column
        lds[dstoff + IDX(i, j + 0)] = v.x;
        lds[dstoff + IDX(i, j + 1)] = v.y;
        lds[dstoff + IDX(i, j + 2)] = v.z;
        lds[dstoff + IDX(i, j + 3)] = v.w;
    }
}

__global__ void k_zero(float* __restrict__ p, int n)
{
    const int i = blockIdx.x * blockDim.x + threadIdx.x;
    if (i < n) p[i] = 0.0f;
}

// ---------------------------------------------------------------------------
// Kernel 1: per-problem kinetic scalars, atomically accumulated into T[b].
//   T_mu    = 0.5 * mu^T Minv mu
//   T_Sigma = tr(pS pS),  pS = pi_Sigma @ Sigma   (WMMA GEMM)
//   T_phi   = 0.5 * |phi|^2
// ---------------------------------------------------------------------------
__global__ __launch_bounds__(NTHREADS) void k_scalars(
    const float* __restrict__ Sigma, const float* __restrict__ piS,
    const float* __restrict__ piMu,  const float* __restrict__ Minv,
    const float* __restrict__ piPhi, float* __restrict__ Tb, int N)
{
    const int OS = 0, OP = MAT_ELEMS, OQ = 2 * MAT_ELEMS;
    __shared__ float red[NTHREADS];
    __shared__ float mu[KDIM];

    const int    p   = blockIdx.x;          // 0 .. B*N-1
    const int    b   = p / N;
    const size_t off = (size_t)p * KDIM * KDIM;

    load_mat(Sigma + off, OS);
    load_mat(piS   + off, OP);
    if (threadIdx.x < KDIM) mu[threadIdx.x] = piMu[(size_t)p * KDIM + threadIdx.x];
    __syncthreads();

    gemm64(OP, OS, OQ, 1.0f, 0.0f);         // Q = pi_Sigma @ Sigma

    float acc = 0.0f;
    for (int idx = threadIdx.x; idx < KDIM * KDIM; idx += NTHREADS) {
        const int i = idx >> 6, j = idx & 63;
        acc += lds[OQ + IDX(i, j)] * lds[OQ + IDX(j, i)];    // tr(pS pS)
        acc += 0.5f * mu[i] * Minv[off + idx] * mu[j];       // T_mu
    }
    if (threadIdx.x == 0) {
        const float* ph = piPhi + (size_t)p * 3;
        acc += 0.5f * (ph[0] * ph[0] + ph[1] * ph[1] + ph[2] * ph[2]);
    }
    red[threadIdx.x] = acc;
    __syncthreads();
    for (int s = NTHREADS / 2; s > 0; s >>= 1) {
        if (threadIdx.x < s) red[threadIdx.x] += red[threadIdx.x + s];
        __syncthreads();
    }
    if (threadIdx.x == 0) atomicAdd(&Tb[b], red[0]);
}

// ---------------------------------------------------------------------------
// Kernel 2: SPD exponential map, eigh-free:
//   S   = sym(Sigma) + eps I,  c = tr(S)/K   (S/c ~ I, NS converges fast)
//   V   = 0.2 * S @ piS @ S
//   NS  : Y->(S/c)^1/2, Z->(S/c)^-1/2        (3 GEMMs/iter)
//   W   = (Z V Z)/c = S^-1/2 V S^-1/2, symmetrized
//   E   = expm(W) via (Taylor6(W/4))^4       (8 GEMMs)
//   out = sym(c * Y E Y) + T[b]
// ---------------------------------------------------------------------------
__global__ __launch_bounds__(NTHREADS) void k_spdmap(
    const float* __restrict__ Sigma, const float* __restrict__ piS,
    const float* __restrict__ Tb, float* __restrict__ out, int N)
{
    const int O0 = 0;
    const int O1 = 1 * MAT_ELEMS;
    const int O2 = 2 * MAT_ELEMS;
    const int O3 = 3 * MAT_ELEMS;
    const int O4 = 4 * MAT_ELEMS;
    const int O5 = 5 * MAT_ELEMS;
    __shared__ float sred[KDIM];
    __shared__ float s_c;

    const int    p   = blockIdx.x;
    const int    b   = p / N;
    const size_t off = (size_t)p * KDIM * KDIM;

    load_mat(Sigma + off, O2);              // raw Sigma
    load_mat(piS   + off, O1);              // pi_Sigma
    __syncthreads();

    // O0 = sym(Sigma) + eps I
    for (int idx = threadIdx.x; idx < KDIM * KDIM; idx += NTHREADS) {
        const int i = idx >> 6, j = idx & 63;
        float v = 0.5f * (lds[O2 + IDX(i, j)] + lds[O2 + IDX(j, i)]);
        if (i == j) v += EPSV;
        lds[O0 + IDX(i, j)] = v;
    }
    __syncthreads();

    // c = tr(S)/K
    if (threadIdx.x < KDIM)
        sred[threadIdx.x] = lds[O0 + IDX(threadIdx.x, threadIdx.x)];
    __syncthreads();
    if (threadIdx.x == 0) {
        float t = 0.0f;
        for (int i = 0; i < KDIM; ++i) t += sred[i];
        s_c = t / (float)KDIM;
    }
    __syncthreads();
    const float c = s_c, rc = 1.0f / c;

    // V = 0.2 * S @ (P @ S)
    gemm64(O1, O0, O2, 1.0f, 0.0f);         // O2 = P @ S
    gemm64(O0, O2, O3, DT2,  0.0f);         // O3 = V

    // Y0 = S/c, Z0 = I
    for (int idx = threadIdx.x; idx < KDIM * KDIM; idx += NTHREADS) {
        const int i = idx >> 6, j = idx & 63;
        lds[O4 + IDX(i, j)] = lds[O0 + IDX(i, j)] * rc;
        lds[O5 + IDX(i, j)] = (i == j) ? 1.0f : 0.0f;
    }
    __syncthreads();

    // Coupled Newton-Schulz (even iteration count -> offsets return home)
    int Y = O4, Z = O5, t0 = O0, t1 = O1, t2 = O2;
    for (int it = 0; it < NS_ITERS; ++it) {
        gemm64(Z,  Y,  t0, -0.5f, 1.5f);    // T = 1.5I - 0.5 Z Y
        gemm64(Y,  t0, t1,  1.0f, 0.0f);    // Y' = Y T
        gemm64(t0, Z,  t2,  1.0f, 0.0f);    // Z' = T Z
        const int oy = Y, oz = Z;
        Y = t1; Z = t2; t1 = oy; t2 = oz;
    }
    // Y = (S/c)^1/2, Z = (S/c)^-1/2  =>  S^1/2 = sqrt(c)Y, S^-1/2 = Z/sqrt(c)

    // W = (Z V Z)/c, then symmetrize into X
    gemm64(Z,  O3, t0, 1.0f, 0.0f);         // t0 = Z V
    gemm64(t0, Z,  t1, rc,   0.0f);         // t1 = W
    const int W = t1;
    const int X = t2;
    for (int idx = threadIdx.x; idx < KDIM * KDIM; idx += NTHREADS) {
        const int i = idx >> 6, j = idx & 63;
        lds[X + IDX(i, j)] = 0.5f * (lds[W + IDX(i, j)] + lds[W + IDX(j, i)]);
    }
    __syncthreads();

    // E = expm(X): order-6 Taylor Horner on X/4, then square twice.
    int E  = t0;
    int Et = t1;
    for (int idx = threadIdx.x; idx < KDIM * KDIM; idx += NTHREADS) {
        const int i = idx >> 6, j = idx & 63;
        lds[E + IDX(i, j)] = (i == j) ? 1.0f : 0.0f;
    }
    __syncthreads();
    for (int k = 6; k >= 1; --k) {          // E <- I + (X/4)/k * E   (6 steps)
        gemm64(X, E, Et, 0.25f / (float)k, 1.0f);
        const int tmp = E; E = Et; Et = tmp;
    }
    gemm64(E,  E,  Et, 1.0f, 0.0f);         // exp(W/2)
    gemm64(Et, Et, E,  1.0f, 0.0f);         // exp(W)

    // Sigma_new = c * Y E Y  (X buffer is free now)
    gemm64(Y,  E, Et, 1.0f, 0.0f);
    gemm64(Et, Y, X,  c,    0.0f);

    // out = sym(Sigma_new) + T[b]
    const float Tval = Tb[b];
    for (int idx = threadIdx.x; idx < KDIM * KDIM; idx += NTHREADS) {
        const int i = idx >> 6, j = idx & 63;
        out[off + idx] = 0.5f * (lds[X + IDX(i, j)] + lds[X + IDX(j, i)]) + Tval;
    }
}

// ---------------------------------------------------------------------------
extern "C" void kernel_launch(void* const* d_in, const int* in_sizes, int n_in,
                              void* d_out, int out_size, void* d_ws, size_t ws_size,
                              hipStream_t stream)
{
    const float* Sigma = (const float*)d_in[0];
    const float* piS   = (const float*)d_in[1];
    const float* piMu  = (const float*)d_in[2];
    const float* Minv  = (const float*)d_in[3];
    const float* piPhi = (const float*)d_in[4];
    float* out = (float*)d_out;
    float* Tb  = (float*)d_ws;               // BATCH floats of scratch

    const int BN = in_sizes[2] / KDIM;       // B*N from pi_mu (B,N,K)
    const int N  = BN / BATCH;

    const size_t lds1 = (size_t)3 * MAT_ELEMS * sizeof(float);  //  ~49.9 KB
    const size_t lds2 = (size_t)6 * MAT_ELEMS * sizeof(float);  //  ~99.8 KB

    k_zero<<<1, 32, 0, stream>>>(Tb, BATCH);
    k_scalars<<<BN, NTHREADS, lds1, stream>>>(Sigma, piS, piMu, Minv, piPhi, Tb, N);
    k_spdmap <<<BN, NTHREADS, lds2, stream>>>(Sigma, piS, Tb, out, N);
}